// GaussianBatchLanguageModel_85718957294351
// MI455X (gfx1250) — compile-verified
//
#include <hip/hip_runtime.h>
#include <hip/hip_bf16.h>
#include <math.h>

#define DIM    8
#define VOCAB  2000
#define BATCH  4
#define SEQ    64
#define LOG2PI 1.8378770664093453f

typedef __attribute__((ext_vector_type(2))) float v2f;
typedef __attribute__((ext_vector_type(8))) float v8f;

// ---------------- workspace layout (float offsets) ----------------
#define WS_LAM_T   0        // 16x16 = 256
#define WS_ETA_T   256      // 16
#define WS_ZETA1   272      // 2000
#define WS_LAM1V   2272     // 2000*8
#define WS_E1      18272    // 2000*8
#define WS_LAM0    34272    // 4*64*64
#define WS_ETA0    50656    // 4*64*8
#define WS_ZETA0   52704    // 4*64
// total 52960 floats (~212 KB)

// ---------------- small dense helpers (fully unrolled, register-resident) ---
__device__ __forceinline__ void chol8(const float A[8][8], float L[8][8]) {
#pragma unroll
  for (int i = 0; i < 8; ++i) {
#pragma unroll
    for (int j = 0; j < 8; ++j) {
      if (j > i) continue;
      float s = A[i][j];
#pragma unroll
      for (int k = 0; k < 8; ++k)
        if (k < j) s -= L[i][k] * L[j][k];
      L[i][j] = (i == j) ? sqrtf(s) : s / L[j][j];
    }
  }
}

__device__ __forceinline__ void solveLLT8(const float L[8][8], const float b[8],
                                          float x[8]) {
  float y[8];
#pragma unroll
  for (int i = 0; i < 8; ++i) {
    float s = b[i];
#pragma unroll
    for (int k = 0; k < 8; ++k)
      if (k < i) s -= L[i][k] * y[k];
    y[i] = s / L[i][i];
  }
#pragma unroll
  for (int ii = 7; ii >= 0; --ii) {
    float s = y[ii];
#pragma unroll
    for (int k = 0; k < 8; ++k)
      if (k > ii) s -= L[k][ii] * x[k];
    x[ii] = s / L[ii][ii];
  }
}

// ---------------- kernel 1: transition prep (uses WMMA f32 16x16x4) ---------
// lam_t = inv(C^T C) = W * W^T, W = inv(C) (C lower-triangular 16x16)
__global__ void prep_trans_kernel(const float* __restrict__ tcho,
                                  const float* __restrict__ tmu,
                                  float* __restrict__ ws) {
  __shared__ float Wsh[16][17];
  __shared__ float Lam[16][17];
  const int tid = threadIdx.x;

  if (tid == 0) {
    // forward substitution: W = C^{-1}, lower triangular
    for (int j = 0; j < 16; ++j) {
      for (int i = 0; i < j; ++i) Wsh[i][j] = 0.0f;
      Wsh[j][j] = 1.0f / tcho[j * 16 + j];
      for (int i = j + 1; i < 16; ++i) {
        float s = 0.0f;
        for (int k = j; k < i; ++k) s += tcho[i * 16 + k] * Wsh[k][j];
        Wsh[i][j] = -s / tcho[i * 16 + i];
      }
    }
  }
  __syncthreads();

  // D = A*B with A = W (16x16), B = W^T (16x16), chained K: 4 x (16x16x4).
  // A layout (32-bit, 16x4): lanes 0-15 -> M=lane, VGPR0=K0,VGPR1=K1;
  //                          lanes 16-31 -> M=lane-16, VGPR0=K2,VGPR1=K3.
  // B layout mirrors with N=lane; B[k][n] = W[n][k], so per-lane a == b.
  const int m  = tid & 15;
  const int hi = tid >> 4;
  v8f acc = {0.f, 0.f, 0.f, 0.f, 0.f, 0.f, 0.f, 0.f};
#pragma unroll
  for (int k0 = 0; k0 < 16; k0 += 4) {
    v2f a, b;
    a.x = Wsh[m][k0 + 2 * hi + 0];
    a.y = Wsh[m][k0 + 2 * hi + 1];
    b.x = Wsh[m][k0 + 2 * hi + 0];  // B[k][n] = W[n][k], n == m per lane
    b.y = Wsh[m][k0 + 2 * hi + 1];
    acc = __builtin_amdgcn_wmma_f32_16x16x4_f32(
        /*neg_a=*/false, a, /*neg_b=*/false, b,
        /*c_mod=*/(short)0, acc, /*reuse_a=*/false, /*reuse_b=*/false);
  }
  // C/D layout: VGPR r -> M = r + 8*hi, N = lane&15
#pragma unroll
  for (int r = 0; r < 8; ++r) Lam[r + 8 * hi][m] = acc[r];
  __syncthreads();

  for (int idx = tid; idx < 256; idx += 32)
    ws[WS_LAM_T + idx] = Lam[idx >> 4][idx & 15];
  if (tid < 16) {
    float s = 0.0f;
    for (int j = 0; j < 16; ++j) s += Lam[tid][j] * tmu[j];
    ws[WS_ETA_T + tid] = s;
  }
}

// ---------------- kernel 2: decoder natural-parameter tables ----------------
__global__ void prep_decoder_kernel(const float* __restrict__ dmu,
                                    const float* __restrict__ dcho,
                                    float* __restrict__ ws) {
  const int v = blockIdx.x * blockDim.x + threadIdx.x;
  if (v >= VOCAB) return;
  float logdet = 0.0f, quad = 0.0f;
#pragma unroll
  for (int k = 0; k < 8; ++k) {
    const float c   = dcho[v * 8 + k];
    const float lam = 1.0f / (c * c);
    const float mu  = dmu[v * 8 + k];
    const float e   = mu * lam;
    ws[WS_LAM1V + v * 8 + k] = lam;
    ws[WS_E1    + v * 8 + k] = e;
    logdet += logf(lam);
    quad   += e * mu;
  }
  ws[WS_ZETA1 + v] = -0.5f * (8.0f * LOG2PI - logdet + quad);
}

// ---------------- kernel 3: information-form Kalman scan --------------------
__global__ void scan_kernel(const int* __restrict__ sentences,
                            const float* __restrict__ emu,
                            const float* __restrict__ echo,
                            float* __restrict__ ws) {
  const int b = threadIdx.x;
  if (b >= BATCH) return;
  const float* lam_t = ws + WS_LAM_T;
  const float* eta_t = ws + WS_ETA_T;

  float T11[8][8], T12[8][8], T21[8][8], T22[8][8];
#pragma unroll
  for (int i = 0; i < 8; ++i)
#pragma unroll
    for (int j = 0; j < 8; ++j) {
      T11[i][j] = lam_t[i * 16 + j];
      T12[i][j] = lam_t[i * 16 + 8 + j];
      T21[i][j] = lam_t[(8 + i) * 16 + j];
      T22[i][j] = lam_t[(8 + i) * 16 + 8 + j];
    }

  float Lam[8][8], eta[8];
#pragma unroll
  for (int i = 0; i < 8; ++i) {
    eta[i] = 0.0f;
#pragma unroll
    for (int j = 0; j < 8; ++j) Lam[i][j] = (i == j) ? 1.0f : 0.0f;
  }

  for (int t = 0; t < SEQ; ++t) {
    const int w = sentences[b * SEQ + t];
    float lw[8], wmu[8];
#pragma unroll
    for (int k = 0; k < 8; ++k) {
      const float c = echo[w * 8 + k];
      lw[k]  = 1.0f / (c * c);
      wmu[k] = emu[w * 8 + k];
    }
    // A = T11 + Lam,  factor
    float A[8][8], L[8][8];
#pragma unroll
    for (int i = 0; i < 8; ++i)
#pragma unroll
      for (int j = 0; j < 8; ++j) A[i][j] = T11[i][j] + Lam[i][j];
    chol8(A, L);

    // X = A^{-1} T12 (column-wise),  xa = A^{-1} (eta_t[:8] + eta)
    float X[8][8];
#pragma unroll
    for (int c = 0; c < 8; ++c) {
      float rhs[8], col[8];
#pragma unroll
      for (int i = 0; i < 8; ++i) rhs[i] = T12[i][c];
      solveLLT8(L, rhs, col);
#pragma unroll
      for (int i = 0; i < 8; ++i) X[i][c] = col[i];
    }
    float a[8], xa[8];
#pragma unroll
    for (int i = 0; i < 8; ++i) a[i] = eta_t[i] + eta[i];
    solveLLT8(L, a, xa);

    // Lam' = T22 + diag(lw) - T21*X ;  eta' = eta_t[8:] + wmu*lw - T21*xa
    float Lp[8][8], ep[8];
#pragma unroll
    for (int i = 0; i < 8; ++i) {
#pragma unroll
      for (int j = 0; j < 8; ++j) {
        float s = T22[i][j];
#pragma unroll
        for (int k = 0; k < 8; ++k) s -= T21[i][k] * X[k][j];
        Lp[i][j] = s;
      }
      Lp[i][i] += lw[i];
      float s2 = eta_t[8 + i];
#pragma unroll
      for (int k = 0; k < 8; ++k) s2 -= T21[i][k] * xa[k];
      ep[i] = s2 + wmu[i] * lw[i];
    }

    const int bt = b * SEQ + t;
#pragma unroll
    for (int i = 0; i < 8; ++i) {
#pragma unroll
      for (int j = 0; j < 8; ++j) ws[WS_LAM0 + bt * 64 + i * 8 + j] = Lp[i][j];
      ws[WS_ETA0 + bt * 8 + i] = ep[i];
    }

    // zeta0 = -0.5*(d*log2pi - logdet(Lam') + eta'^T Lam'^{-1} eta')
    float L2[8][8];
    chol8(Lp, L2);
    float logdet = 0.0f;
#pragma unroll
    for (int i = 0; i < 8; ++i) logdet += logf(fabsf(L2[i][i]));
    logdet *= 2.0f;
    float m2[8];
    solveLLT8(L2, ep, m2);
    float quad = 0.0f;
#pragma unroll
    for (int i = 0; i < 8; ++i) quad += ep[i] * m2[i];
    ws[WS_ZETA0 + bt] = -0.5f * (8.0f * LOG2PI - logdet + quad);

    // carry in information form (next lam1 == Lam', next eta1 == eta')
#pragma unroll
    for (int i = 0; i < 8; ++i) {
      eta[i] = ep[i];
#pragma unroll
      for (int j = 0; j < 8; ++j) Lam[i][j] = Lp[i][j];
    }
  }
}

// ---------------- kernel 4: bulk scoring (hot kernel) -----------------------
// Block-shared lam0/eta0/zeta0 are staged global->LDS with the CDNA5 async
// data mover path (GLOBAL_LOAD_ASYNC_TO_LDS_B32, ASYNCcnt-tracked), then each
// wave drains its ASYNCcnt before the block barrier.
__global__ void score_kernel(const float* __restrict__ ws,
                             float* __restrict__ out) {
  const int bt  = blockIdx.y;                          // b*SEQ + t
  const int tid = threadIdx.x;
  const int v   = blockIdx.x * blockDim.x + tid;

  __shared__ float lam0s[64];
  __shared__ float eta0s[8];
  __shared__ float zeta0s[1];

  if (v < VOCAB) {
    __builtin_prefetch((const void*)&ws[WS_LAM1V + (size_t)v * 8], 0, 0);
    __builtin_prefetch((const void*)&ws[WS_E1 + (size_t)v * 8], 0, 0);
  }

  // --- async global -> LDS stage of the 73 block-shared floats ---
  if (tid < 73) {
    const float* g;
    float* l;
    if (tid < 64) {
      g = &ws[WS_LAM0 + bt * 64 + tid];
      l = &lam0s[tid];
    } else if (tid < 72) {
      g = &ws[WS_ETA0 + bt * 8 + (tid - 64)];
      l = &eta0s[tid - 64];
    } else {
      g = &ws[WS_ZETA0 + bt];
      l = &zeta0s[0];
    }
    // generic pointer to __shared__ is {aperture_hi32, lds_offset_lo32};
    // hardware uses addr[31:0] as the wave-relative LDS byte address.
    unsigned int lds_off = (unsigned int)(unsigned long long)(void*)l;
    asm volatile("global_load_async_to_lds_b32 %0, %1, off"
                 :
                 : "v"(lds_off), "v"(g)
                 : "memory");
  }
  asm volatile("s_wait_asynccnt 0x0" ::: "memory");
  __syncthreads();
  if (v >= VOCAB) return;

  float lw[8], e[8];
#pragma unroll
  for (int k = 0; k < 8; ++k) {
    lw[k] = ws[WS_LAM1V + v * 8 + k];
    e[k]  = eta0s[k] + ws[WS_E1 + v * 8 + k];
  }

  // Cholesky of lam0 + diag(lw)  (lower triangle, registers)
  float L[8][8];
#pragma unroll
  for (int i = 0; i < 8; ++i)
#pragma unroll
    for (int j = 0; j < 8; ++j) {
      if (j > i) continue;
      float s = lam0s[i * 8 + j] + ((i == j) ? lw[i] : 0.0f);
#pragma unroll
      for (int k = 0; k < 8; ++k)
        if (k < j) s -= L[i][k] * L[j][k];
      L[i][j] = (i == j) ? sqrtf(s) : s / L[j][j];
    }

  float logdet = 0.0f;
#pragma unroll
  for (int i = 0; i < 8; ++i) logdet += logf(L[i][i]);
  logdet *= 2.0f;

  // z = L^{-1} e ; part3 = |z|^2
  float z[8], part3 = 0.0f;
#pragma unroll
  for (int i = 0; i < 8; ++i) {
    float s = e[i];
#pragma unroll
    for (int k = 0; k < 8; ++k)
      if (k < i) s -= L[i][k] * z[k];
    z[i] = s / L[i][i];
    part3 += z[i] * z[i];
  }
  const float zeta_n = -0.5f * (8.0f * LOG2PI - logdet + part3);
  out[(size_t)bt * VOCAB + v] = zeta0s[0] + ws[WS_ZETA1 + v] - zeta_n;
}

// ---------------- launcher --------------------------------------------------
extern "C" void kernel_launch(void* const* d_in, const int* in_sizes, int n_in,
                              void* d_out, int out_size, void* d_ws,
                              size_t ws_size, hipStream_t stream) {
  const int*   sentences = (const int*)d_in[0];
  // d_in[1] = masks (unused by the reference computation)
  const float* emu  = (const float*)d_in[2];   // emission_mu_emb  (2000x8)
  const float* echo = (const float*)d_in[3];   // emission_cho_emb (2000x8)
  const float* tmu  = (const float*)d_in[4];   // transition_mu    (16)
  const float* tcho = (const float*)d_in[5];   // transition_cho   (16x16)
  const float* dmu  = (const float*)d_in[6];   // decoder_mu       (2000x8)
  const float* dcho = (const float*)d_in[7];   // decoder_cho      (2000x8)
  float* ws  = (float*)d_ws;
  float* out = (float*)d_out;

  prep_trans_kernel<<<1, 32, 0, stream>>>(tcho, tmu, ws);
  prep_decoder_kernel<<<(VOCAB + 255) / 256, 256, 0, stream>>>(dmu, dcho, ws);
  scan_kernel<<<1, 32, 0, stream>>>(sentences, emu, echo, ws);
  dim3 grid((VOCAB + 255) / 256, BATCH * SEQ);
  score_kernel<<<grid, 256, 0, stream>>>(ws, out);
}